// RNNCell_62122406969624
// MI455X (gfx1250) — compile-verified
//
#include <hip/hip_runtime.h>

typedef __attribute__((ext_vector_type(2))) float v2f;
typedef __attribute__((ext_vector_type(8))) float v8f;

#define T_STEPS 512
#define BATCH   16384
#define H       32

#define LOG2E_F  1.4426950408889634f
#define LN2_F    0.6931471805599453f

// sigmoid(z) = 0.5*tanh(z/2) + 0.5  -- exact rewrite; uses CDNA5 v_tanh_f32
// (1 trans + 2 VALU instead of exp+rcp+select: 2 trans + 5 VALU).
__device__ __forceinline__ float sigmoidf(float z) {
#if __has_builtin(__builtin_amdgcn_tanhf)
  return fmaf(0.5f, __builtin_amdgcn_tanhf(0.5f * z), 0.5f);
#else
  const float q = __builtin_amdgcn_exp2f(-fabsf(z) * LOG2E_F);
  const float r = __builtin_amdgcn_rcpf(1.0f + q);
  return (z >= 0.0f) ? r : q * r;
#endif
}

// softplus via raw exp2/log2 (arguments provably in safe ranges: exp2 arg <= 0,
// log2 arg in [1,2], so no libm range fixups); sigmoid via tanh.
__device__ __forceinline__ void softplus_sigmoid(float z, float& sp, float& sg) {
  const float q = __builtin_amdgcn_exp2f(-fabsf(z) * LOG2E_F);   // e^{-|z|}
  sp = fmaxf(z, 0.0f) + LN2_F * __builtin_amdgcn_logf(1.0f + q); // softplus(z)
  sg = sigmoidf(z);                                              // sigmoid(z)
}

// lane <-> lane^16 exchange: ds_swizzle SWAPX16 (xor=0x10, or=0, and=0x1f)
__device__ __forceinline__ float swap16(float x) {
  return __int_as_float(__builtin_amdgcn_ds_swizzle(__float_as_int(x), 0x401F));
}

// Convert a 32(hidden) x 16(batch) column group from WMMA C/D layout
// (lane half hi holds hidden rows {m*16 + r + 8*hi}) to f32 B-operand layout
// (lane half hi holds hidden rows {kt*4 + 2*hi + v}). One half-wave exchange.
__device__ __forceinline__ void cToB(const float (&C)[16], float (&Bv)[16], int hi) {
  float sw[16];
#pragma unroll
  for (int j = 0; j < 16; ++j) sw[j] = swap16(C[j]);
#pragma unroll
  for (int kt = 0; kt < 8; ++kt) {
#pragma unroll
    for (int v = 0; v < 2; ++v) {
      const int h0 = 4 * kt + v;                   // row wanted by hi==0 lanes
      const int j0 = (h0 & 7) + 8 * (h0 >> 4);
      const float val0 = ((h0 >> 3) & 1) ? sw[j0] : C[j0];
      const int h1 = 4 * kt + 2 + v;               // row wanted by hi==1 lanes
      const int j1 = (h1 & 7) + 8 * (h1 >> 4);
      const float val1 = ((h1 >> 3) & 1) ? C[j1] : sw[j1];
      Bv[kt * 2 + v] = hi ? val1 : val0;
    }
  }
}

__global__ __launch_bounds__(32)
void rnn_scan_kernel(const float* __restrict__ eps, const float* __restrict__ hs,
                     const float* __restrict__ W1,  const float* __restrict__ b1,
                     const float* __restrict__ W2,  const float* __restrict__ b2,
                     const float* __restrict__ W3,  float* __restrict__ sig) {
  const int lane = threadIdx.x & 31;
  const int p    = lane & 15;          // batch column within the 16-wide N tile
  const int hi   = (lane >> 4) & 1;    // half-wave id (selects hidden-row subset)
  const int wave = blockIdx.x;         // one wave per block
  const int base = wave * 16;          // 16 batch elements per wave

  // ---- loop-invariant per-lane constants in C/D layout (hidden h = m*16 + r + 8*hi)
  float w10C[16], w11C[16], b1C[16], b2C[16], w3C[16];
#pragma unroll
  for (int m = 0; m < 2; ++m) {
#pragma unroll
    for (int r = 0; r < 8; ++r) {
      const int j = m * 8 + r;
      const int h = m * 16 + r + hi * 8;
      w10C[j] = W1[0 * H + h];
      w11C[j] = W1[1 * H + h];
      b1C[j]  = b1[h];
      b2C[j]  = b2[h];
      w3C[j]  = W3[h];
    }
  }

  // ---- loop-invariant A-operand tiles (f32 WMMA A layout: M = lane&15,
  //      K = kt*4 + 2*hi + v). A1 = W2^T (for z2^T), A2 = W2 (for gsum^T).
  v2f A1[2][8], A2[2][8];
#pragma unroll
  for (int mt = 0; mt < 2; ++mt) {
#pragma unroll
    for (int kt = 0; kt < 8; ++kt) {
      const int k0  = kt * 4 + hi * 2;
      const int col = mt * 16 + p;
      v2f a1t, a2t;
      a1t.x = W2[(k0 + 0) * H + col];  a1t.y = W2[(k0 + 1) * H + col];
      a2t.x = W2[col * H + (k0 + 0)];  a2t.y = W2[col * H + (k0 + 1)];
      A1[mt][kt] = a1t;
      A2[mt][kt] = a2t;
    }
  }

  // gamma state for this lane's batch (base+p); redundant across the
  // half-wave pair but kept consistent by identical arithmetic.
  float gA = 0.0f;

  const float* ep = eps + base;
  const float* hp = hs + base;
  float*       op = sig + base;

  for (int t = 0; t < T_STEPS; ++t) {
    const float eA  = ep[p];
    const float hAv = hp[p];
    // speculative prefetch of next timestep (dropped if OOB)
    __builtin_prefetch(ep + BATCH, 0, 0);
    __builtin_prefetch(hp + BATCH, 0, 0);

    // ---- layer 1 (K=2, pure VALU) in C/D layout; keep sigmoid(z1) for later
    float a1C[16], s1C[16];
#pragma unroll
    for (int j = 0; j < 16; ++j) {
      const float z = fmaf(eA, w10C[j], fmaf(gA, w11C[j], b1C[j]));
      softplus_sigmoid(z, a1C[j], s1C[j]);
    }

    // ---- relayout a1 to B-operand form
    float a1B[16];
    cToB(a1C, a1B, hi);

    // ---- GEMM1: z2^T = W2^T @ a1^T + b2   (16 x v_wmma_f32_16x16x4_f32)
    v8f acc1[2];
#pragma unroll
    for (int mt = 0; mt < 2; ++mt) {
      v8f c;
#pragma unroll
      for (int r = 0; r < 8; ++r) c[r] = b2C[mt * 8 + r];
#pragma unroll
      for (int kt = 0; kt < 8; ++kt) {
        v2f bop;
        bop.x = a1B[kt * 2 + 0];
        bop.y = a1B[kt * 2 + 1];
        c = __builtin_amdgcn_wmma_f32_16x16x4_f32(
                false, A1[mt][kt], false, bop, (short)0, c, false, false);
      }
      acc1[mt] = c;
    }

    // ---- v2 = sigmoid(z2) * W3   (elementwise, C/D layout)
    float v2C[16];
#pragma unroll
    for (int mt = 0; mt < 2; ++mt)
#pragma unroll
      for (int r = 0; r < 8; ++r) {
        const int j = mt * 8 + r;
        v2C[j] = sigmoidf(acc1[mt][r]) * w3C[j];
      }

    float v2B[16];
    cToB(v2C, v2B, hi);

    // ---- GEMM2: gsum^T = W2 @ v2^T   (16 x v_wmma_f32_16x16x4_f32)
    v8f acc2[2];
#pragma unroll
    for (int mt = 0; mt < 2; ++mt) {
      v8f c = {0.f, 0.f, 0.f, 0.f, 0.f, 0.f, 0.f, 0.f};
#pragma unroll
      for (int kt = 0; kt < 8; ++kt) {
        v2f bop;
        bop.x = v2B[kt * 2 + 0];
        bop.y = v2B[kt * 2 + 1];
        c = __builtin_amdgcn_wmma_f32_16x16x4_f32(
                false, A2[mt][kt], false, bop, (short)0, c, false, false);
      }
      acc2[mt] = c;
    }

    // ---- v1 = sigmoid(z1) * gsum;  g = v1 @ W1^T  (per-lane partial + half-wave add)
    float s0 = 0.0f, s1 = 0.0f;
#pragma unroll
    for (int mt = 0; mt < 2; ++mt)
#pragma unroll
      for (int r = 0; r < 8; ++r) {
        const int j = mt * 8 + r;
        const float v1 = s1C[j] * acc2[mt][r];
        s0 = fmaf(v1, w10C[j], s0);
        s1 = fmaf(v1, w11C[j], s1);
      }
    s0 += swap16(s0);   // sig_n  (d x / d eps): full 32-row sum, both halves
    s1 += swap16(s1);   // gs     (d x / d gamma)

    // ---- gamma <- gamma - hs * gs ; emit sig_n (coalesced b32 store, low half)
    gA = fmaf(-hAv, s1, gA);
    if (lane < 16) op[p] = s0;

    ep += BATCH; hp += BATCH; op += BATCH;
  }
}

extern "C" void kernel_launch(void* const* d_in, const int* in_sizes, int n_in,
                              void* d_out, int out_size, void* d_ws, size_t ws_size,
                              hipStream_t stream) {
  const float* eps = (const float*)d_in[0];
  const float* hs  = (const float*)d_in[1];
  const float* W1  = (const float*)d_in[2];
  const float* b1  = (const float*)d_in[3];
  const float* W2  = (const float*)d_in[4];
  const float* b2  = (const float*)d_in[5];
  const float* W3  = (const float*)d_in[6];
  // d_in[7] = b3: unused (cancels in the analytic gradients)
  float* out = (float*)d_out;

  const int waves = BATCH / 16;   // 1024 single-wave workgroups, spread over SIMDs
  rnn_scan_kernel<<<waves, 32, 0, stream>>>(eps, hs, W1, b1, W2, b2, W3, out);
}